// ComponentWiseSpline_7602092114224
// MI455X (gfx1250) — compile-verified
//
#include <hip/hip_runtime.h>
#include <stdint.h>

#define KBINS  32
#define DIMS   256
#define ROWS   16
#define BOUNDF 3.0f
#define MINW   1e-3f
#define MINH   1e-3f
#define MINDER 1e-3f
#define MINLAM 0.025f
#define EPSF   1e-6f
#define REC    12   // padded per-bin record stride in floats (48 bytes, 16B aligned)
#define LN2F   0.69314718055994530942f

#if __has_builtin(__builtin_amdgcn_wmma_f32_16x16x4_f32)
#define USE_WMMA_REDUCE 1
typedef float v2f __attribute__((ext_vector_type(2)));
typedef float v8f __attribute__((ext_vector_type(8)));
#endif

#if __has_builtin(__builtin_amdgcn_global_load_async_to_lds_b128)
#define USE_ASYNC_COPY 1
#define AS1 __attribute__((address_space(1)))
#define AS3 __attribute__((address_space(3)))
typedef int v4i __attribute__((vector_size(4 * sizeof(int))));
#endif

// ---------------- wave32 helpers ----------------
__device__ __forceinline__ float wmax32(float v) {
#pragma unroll
  for (int o = 16; o; o >>= 1) v = fmaxf(v, __shfl_xor(v, o, 32));
  return v;
}
__device__ __forceinline__ float wsum32(float v) {
#pragma unroll
  for (int o = 16; o; o >>= 1) v += __shfl_xor(v, o, 32);
  return v;
}
__device__ __forceinline__ float wscan32(float v, int lane) {  // inclusive prefix sum
#pragma unroll
  for (int o = 1; o < 32; o <<= 1) {
    float t = __shfl_up(v, o, 32);
    if (lane >= o) v += t;
  }
  return v;
}

// softmax -> bin sizes -> knot interval [lo,hi] for this lane's bin (matches reference _knots)
__device__ __forceinline__ void knots32(float raw, int k, float min_size,
                                        float& lo, float& hi) {
  float m    = wmax32(raw);
  float e    = expf(raw - m);
  float s    = wsum32(e);
  float frac = e / s;
  float sz   = min_size + (1.0f - min_size * (float)KBINS) * frac;
  float c    = wscan32(sz, k);          // inclusive cumsum
  float ce   = __shfl_up(c, 1, 32);     // exclusive cumsum (lane0 garbage, overridden)
  lo = (k == 0)         ? -BOUNDF : fmaf(2.0f * BOUNDF, ce, -BOUNDF);
  hi = (k == KBINS - 1) ?  BOUNDF : fmaf(2.0f * BOUNDF, c,  -BOUNDF);
}

// ---------------- kernel 1: derive per-bin spline records (D blocks x 32 lanes) ----------------
__global__ __launch_bounds__(32) void spline_precompute_kernel(
    const float* __restrict__ uw, const float* __restrict__ uh,
    const float* __restrict__ ud, const float* __restrict__ ul,
    float* __restrict__ ws_search, float* __restrict__ ws_params) {
  const int d = blockIdx.x;
  const int k = threadIdx.x;

  float cw_lo, cw_hi, ch_lo, ch_hi;
  knots32(uw[d * KBINS + k], k, MINW, cw_lo, cw_hi);
  knots32(uh[d * KBINS + k], k, MINH, ch_lo, ch_hi);
  const float ww = cw_hi - cw_lo;
  const float hh = ch_hi - ch_lo;
  const float ya = ch_lo;
  const float yb = ch_hi;

  // derivatives at knots: derivs[0]=derivs[K]=1, interior = MINDER + softplus(ud)
  float udv = (k < KBINS - 1) ? ud[d * (KBINS - 1) + k] : 0.0f;
  float sp  = fmaxf(udv, 0.0f) + log1pf(expf(-fabsf(udv)));  // stable softplus
  float dv  = MINDER + sp;
  float d1  = (k == KBINS - 1) ? 1.0f : dv;      // derivs[k+1]
  float dp  = __shfl_up(d1, 1, 32);
  float d0  = (k == 0) ? 1.0f : dp;              // derivs[k]

  // lambda
  float lv = ul[d * KBINS + k];
  float il = MINLAM + (1.0f - 2.0f * MINLAM) * (1.0f / (1.0f + expf(-lv)));

  // derived rational-spline quantities (wa == 1)
  float wb     = sqrtf(d0 / d1);
  float idelta = hh / ww;
  float wc     = (il * d0 + (1.0f - il) * wb * d1) / idelta;
  float yc     = ((1.0f - il) * ya + il * wb * yb) / ((1.0f - il) + il * wb);
  float inv_w  = 1.0f / ww;
  float dn_lo  = wc * il * (yc - ya) * inv_w;                // dnum, low branch
  float dn_hi  = wb * wc * (1.0f - il) * (yb - yc) * inv_w;  // dnum, high branch

  // transposed search table (bank-conflict-free indexing by d in the main kernel)
  ws_search[k * DIMS + d] = cw_lo + EPSF;

  float* rec = ws_params + ((size_t)(d * KBINS + k)) * REC;
  rec[0] = cw_lo;  rec[1] = inv_w;   rec[2] = il;  rec[3] = ya;
  rec[4] = wc;     rec[5] = wc * yc; rec[6] = wb;  rec[7] = wb * yb;
  rec[8] = dn_lo;  rec[9] = dn_hi;
}

// ---------------- kernel 2: streaming spline evaluation ----------------
__global__ __launch_bounds__(256) void spline_main_kernel(
    const float* __restrict__ x,
    const float* __restrict__ ws_search, const float* __restrict__ ws_params,
    float* __restrict__ u_out, float* __restrict__ ld_out) {
  __shared__ __align__(16) float lds_search[KBINS * DIMS];  // 32 KB, layout [bin][dim]
  __shared__ float lds_part[ROWS * 8];

  const int t = threadIdx.x;

  // ---- stage bucketize table global -> LDS (CDNA5 async-to-LDS path) ----
#if defined(USE_ASYNC_COPY)
  {
    unsigned long long g = (unsigned long long)(size_t)ws_search;
    unsigned int lbase   = (unsigned int)(size_t)(&lds_search[0]);  // flat low 32 bits == LDS offset
#pragma unroll
    for (int off = 0; off < (int)sizeof(lds_search); off += 256 * 16) {
      unsigned int o = (unsigned int)off + (unsigned int)t * 16u;
      __builtin_amdgcn_global_load_async_to_lds_b128(
          reinterpret_cast<AS1 v4i*>(g + (unsigned long long)o),
          reinterpret_cast<AS3 v4i*>(lbase + o),
          0, 0);
    }
#if __has_builtin(__builtin_amdgcn_s_wait_asynccnt)
    __builtin_amdgcn_s_wait_asynccnt(0);
#else
    asm volatile("s_wait_asynccnt 0" ::: "memory");
#endif
  }
#else
  for (int i = t; i < KBINS * DIMS; i += 256) lds_search[i] = ws_search[i];
#endif
  __syncthreads();

  const int d       = t;            // one thread per feature dim
  const int wv      = t >> 5;
  const int lane    = t & 31;
  const float* prec = ws_params + (size_t)d * KBINS * REC;
  const size_t base = (size_t)blockIdx.x * ROWS * DIMS;

  for (int r = 0; r < ROWS; ++r) {
    // x is streamed exactly once: non-temporal load keeps L2 free for the param tables
    float xv = __builtin_nontemporal_load(&x[base + (size_t)r * DIMS + d]);

    float xc = fminf(fmaxf(xv, -BOUNDF), BOUNDF);

    // 5-step binary search for bin index (largest b with knot[b]+EPS <= xc, clipped)
    int b = (lds_search[16 * DIMS + d] <= xc) ? 16 : 0;
    b += (lds_search[(b + 8) * DIMS + d] <= xc) ? 8 : 0;
    b += (lds_search[(b + 4) * DIMS + d] <= xc) ? 4 : 0;
    b += (lds_search[(b + 2) * DIMS + d] <= xc) ? 2 : 0;
    b += (lds_search[(b + 1) * DIMS + d] <= xc) ? 1 : 0;

    const float* rp = prec + b * REC;
    float4 p0 = *reinterpret_cast<const float4*>(rp);      // icw, 1/iw, lambda, ya
    float4 p1 = *reinterpret_cast<const float4*>(rp + 4);  // wc, wc*yc, wb, wb*yb
    float2 p2 = *reinterpret_cast<const float2*>(rp + 8);  // dnum_lo, dnum_hi

    float theta = (xc - p0.x) * p0.y;
    float il = p0.z, ya = p0.w, wc = p1.x, wcyc = p1.y, wb = p1.z, wbyb = p1.w;

    bool  lo  = theta <= il;
    float s1  = lo ? (il - theta) : (1.0f - theta);
    float s2  = lo ? theta        : (theta - il);
    float num = (lo ? ya   : wcyc) * s1 + (lo ? wcyc : wbyb) * s2;
    float den = (lo ? 1.0f : wc)   * s1 + (lo ? wc   : wb)   * s2;
    float dn  = lo ? p2.x : p2.y;

#if __has_builtin(__builtin_amdgcn_rcpf)
    float invd = __builtin_amdgcn_rcpf(den);
#else
    float invd = 1.0f / den;
#endif
    bool  inside = (xv >= -BOUNDF) && (xv <= BOUNDF);
    float uu = inside ? num * invd : xv;

    // logabsdet = log(dnum) - 2*log(den) = ln2 * log2(dnum * invd^2)
    // dnum>0 and den>0 inside the box, so raw v_log_f32 (log2, ~1 ulp) is safe & fast.
#if __has_builtin(__builtin_amdgcn_logf)
    float lg2 = __builtin_amdgcn_logf(dn * invd * invd);
#else
    float lg2 = __log2f(dn * invd * invd);
#endif
    float ldv = inside ? LN2F * lg2 : 0.0f;

    __builtin_nontemporal_store(uu, &u_out[base + (size_t)r * DIMS + d]);

    // ---- cross-lane (wave32) sum of ldv ----
    float wsum;
#if defined(USE_WMMA_REDUCE)
    {
      // D = A(16x4) * ones(4x16): every column of D holds row-sums of A, so summing
      // all 16 M-rows of D (8 VGPRs + one lane16 fold) yields the total of all A
      // elements = the 32-lane sum — independent of the exact A lane striping.
      v2f A; A.x = ldv; A.y = 0.0f;
      v2f Bm; Bm.x = 1.0f; Bm.y = 1.0f;
      v8f C = {};
      C = __builtin_amdgcn_wmma_f32_16x16x4_f32(false, A, false, Bm,
                                                (short)0, C, false, false);
      wsum = ((C[0] + C[1]) + (C[2] + C[3])) + ((C[4] + C[5]) + (C[6] + C[7]));
      wsum += __shfl_xor(wsum, 16, 32);
    }
#else
    wsum = ldv;
#pragma unroll
    for (int o = 16; o; o >>= 1) wsum += __shfl_xor(wsum, o, 32);
#endif
    if (lane == 0) lds_part[r * 8 + wv] = wsum;
  }
  __syncthreads();

  // combine the 8 wave partials per row
  if (t < ROWS) {
    float s = 0.0f;
#pragma unroll
    for (int w = 0; w < 8; ++w) s += lds_part[t * 8 + w];
    ld_out[(size_t)blockIdx.x * ROWS + t] = s;
  }
}

extern "C" void kernel_launch(void* const* d_in, const int* in_sizes, int n_in,
                              void* d_out, int out_size, void* d_ws, size_t ws_size,
                              hipStream_t stream) {
  const float* x  = (const float*)d_in[0];
  const float* uw = (const float*)d_in[1];
  const float* uh = (const float*)d_in[2];
  const float* ud = (const float*)d_in[3];
  const float* ul = (const float*)d_in[4];

  const int B = in_sizes[0] / DIMS;  // 32768

  float* out    = (float*)d_out;
  float* u_out  = out;                       // [B, D]
  float* ld_out = out + (size_t)B * DIMS;    // [B]

  float* ws_search = (float*)d_ws;                 // KBINS*DIMS floats (32 KB)
  float* ws_params = ws_search + KBINS * DIMS;     // D*K*REC floats (384 KB)

  spline_precompute_kernel<<<DIMS, 32, 0, stream>>>(uw, uh, ud, ul, ws_search, ws_params);
  spline_main_kernel<<<B / ROWS, 256, 0, stream>>>(x, ws_search, ws_params, u_out, ld_out);
}